// ContrastiveNodeEdgePrompt_34248069218341
// MI455X (gfx1250) — compile-verified
//
#include <hip/hip_runtime.h>
#include <hip/hip_bf16.h>
#include <stdint.h>

typedef __attribute__((ext_vector_type(2))) float v2f;
typedef __attribute__((ext_vector_type(8))) float v8f;

#define DIMD 128
#define NA 5

static __device__ inline v8f wmma_f32_4(v2f a, v2f b, v8f c) {
    // V_WMMA_F32_16X16X4_F32 : D = A(16x4) * B(4x16) + C(16x16), full fp32
    return __builtin_amdgcn_wmma_f32_16x16x4_f32(
        false, a, false, b, (short)0, c, false, false);
}

// CDNA5 async global->LDS copy (16B per lane), tracked on ASYNCcnt.
static __device__ inline void async_copy_b128(uint32_t lds_dst, const void* gaddr) {
    asm volatile("global_load_async_to_lds_b128 %0, %1, off"
                 :: "v"(lds_dst), "v"((uint64_t)(uintptr_t)gaddr)
                 : "memory");
}
static __device__ inline void async_wait0() {
    asm volatile("s_wait_asynccnt 0x0" ::: "memory");
}

// ---------------------------------------------------------------------------
// Kernel 1: out[i] = x[i] + softmax(x @ attn_w^T + attn_b) @ node_anchor
// block = 128 threads (4 waves), 16 nodes per wave, 64 nodes per block
// ---------------------------------------------------------------------------
__global__ void node_prompt_kernel(const float* __restrict__ x,
                                   const float* __restrict__ attn_w,
                                   const float* __restrict__ attn_b,
                                   const float* __restrict__ node_anchor,
                                   float* __restrict__ out, int N) {
    __shared__ float sx[64 * DIMD];      // x tile
    __shared__ float sw[16 * DIMD];      // attn_w zero-padded to 16 rows
    __shared__ float sa[8 * DIMD];       // node_anchor zero-padded to 8 rows
    __shared__ float swt[4][16][18];     // logits -> softmax weights (transposed)
    __shared__ float sbias[8];

    const int t    = threadIdx.x;
    const int wave = t >> 5;
    const int lane = t & 31;
    const int half = lane >> 4;
    const int n    = lane & 15;
    const int base = blockIdx.x * 64;

    // stage zero-padded weights (unconditional B fragments later)
    for (int i = t; i < 16 * DIMD; i += 128) sw[i] = (i < NA * DIMD) ? attn_w[i] : 0.0f;
    for (int i = t; i < 8 * DIMD; i += 128)  sa[i] = (i < NA * DIMD) ? node_anchor[i] : 0.0f;
    if (t < 8) sbias[t] = (t < NA) ? attn_b[t] : 0.0f;

    // stage 64 x-rows via async global->LDS (clamped ragged tail)
    for (int idx = t; idx < 64 * (DIMD / 4); idx += 128) {
        int r = idx >> 5, c4 = idx & 31;
        int gr = min(base + r, N - 1);
        async_copy_b128((uint32_t)(uintptr_t)&sx[r * DIMD + c4 * 4],
                        &x[(size_t)gr * DIMD + c4 * 4]);
    }
    async_wait0();
    __syncthreads();

    // GEMM1: logits[16x16] = x_tile[16x128] @ attn_w^T   (K = 128 -> 32 wmma)
    const int m = wave * 16 + n;
    v8f c = {};
    #pragma unroll 8
    for (int ks = 0; ks < DIMD / 4; ++ks) {
        int k0 = 4 * ks + 2 * half;
        v2f a = *(const v2f*)&sx[m * DIMD + k0];
        v2f b = *(const v2f*)&sw[n * DIMD + k0];
        c = wmma_f32_4(a, b, c);
    }

    // dump raw logits into the transpose buffer
    #pragma unroll
    for (int v = 0; v < 8; ++v)
        swt[wave][v + 8 * half][n] = c[v];
    __syncthreads();

    // serial 5-wide softmax, one lane per row; zero-pad K = 5..7 for GEMM2
    if (lane < 16) {
        float L[NA], mx = -3.0e38f;
        #pragma unroll
        for (int k = 0; k < NA; ++k) {
            L[k] = swt[wave][lane][k] + sbias[k];
            mx = fmaxf(mx, L[k]);
        }
        float s = 0.0f;
        #pragma unroll
        for (int k = 0; k < NA; ++k) { L[k] = __expf(L[k] - mx); s += L[k]; }
        float inv = 1.0f / s;
        #pragma unroll
        for (int k = 0; k < NA; ++k) swt[wave][lane][k] = L[k] * inv;
        #pragma unroll
        for (int k = NA; k < 8; ++k) swt[wave][lane][k] = 0.0f;
    }
    __syncthreads();

    // GEMM2: prompt[16x128] = w[16x8pad] @ node_anchor ; out = x + prompt
    #pragma unroll
    for (int t8 = 0; t8 < DIMD / 16; ++t8) {
        v8f d = {};
        #pragma unroll
        for (int ks = 0; ks < 2; ++ks) {
            int k0 = 4 * ks + 2 * half;
            v2f a = *(const v2f*)&swt[wave][n][k0];
            v2f b;
            b.x = sa[(k0    ) * DIMD + t8 * 16 + n];
            b.y = sa[(k0 + 1) * DIMD + t8 * 16 + n];
            d = wmma_f32_4(a, b, d);
        }
        #pragma unroll
        for (int v = 0; v < 8; ++v) {
            int mr = v + 8 * half;
            int gr = base + wave * 16 + mr;
            if (gr < N)
                out[(size_t)gr * DIMD + t8 * 16 + n] =
                    sx[(wave * 16 + mr) * DIMD + t8 * 16 + n] + d[v];
        }
    }
}

// ---------------------------------------------------------------------------
// Kernel 2: per edge: b = softmax(lrelu([x_src,x_dst] @ edge_w^T + edge_b))
//           atomically out[src] += 0.5*b@edge_anchor ; out[dst] += same
// block = 64 threads (2 waves), 16 edges per wave, 32 edges per block
// ---------------------------------------------------------------------------
__global__ void edge_prompt_kernel(const float* __restrict__ x,
                                   const float* __restrict__ edge_w,
                                   const float* __restrict__ edge_b,
                                   const float* __restrict__ edge_anchor,
                                   const int* __restrict__ eidx,   // int64 viewed as int pairs
                                   float* __restrict__ out, int E) {
    __shared__ float comb[32 * 2 * DIMD];   // [edge][256] concatenated rows
    __shared__ float sew[16 * 2 * DIMD];    // edge_w zero-padded to 16 rows
    __shared__ float sea[8 * DIMD];         // edge_anchor zero-padded to 8 rows
    __shared__ float swt[2][16][18];
    __shared__ float sbias[8];
    __shared__ int   sid[32], did[32];

    const int t    = threadIdx.x;
    const int wave = t >> 5;
    const int lane = t & 31;
    const int half = lane >> 4;
    const int n    = lane & 15;
    const int ebase = blockIdx.x * 32;

    for (int i = t; i < 16 * 2 * DIMD; i += 64) sew[i] = (i < NA * 2 * DIMD) ? edge_w[i] : 0.0f;
    for (int i = t; i < 8 * DIMD; i += 64)      sea[i] = (i < NA * DIMD) ? edge_anchor[i] : 0.0f;
    if (t < 8) sbias[t] = (t < NA) ? edge_b[t] : 0.0f;
    if (t < 32) {
        int e = min(ebase + t, E - 1);
        sid[t] = eidx[2 * e];            // low dword of int64 src id
        did[t] = eidx[2 * (E + e)];      // low dword of int64 dst id
    }
    __syncthreads();

    // gather endpoint rows (64 rows x 512B) via async global->LDS
    for (int j = 0; j < 32; ++j) {
        int r  = wave * 32 + j;
        int el = r >> 1, h = r & 1;
        int id = h ? did[el] : sid[el];
        async_copy_b128((uint32_t)(uintptr_t)&comb[el * 2 * DIMD + h * DIMD + lane * 4],
                        &x[(size_t)id * DIMD + lane * 4]);
    }
    async_wait0();
    __syncthreads();

    // GEMM1: logits[16x16] = comb[16x256] @ edge_w^T   (K = 256 -> 64 wmma)
    const int m = wave * 16 + n;
    v8f c = {};
    #pragma unroll 8
    for (int ks = 0; ks < 2 * DIMD / 4; ++ks) {
        int k0 = 4 * ks + 2 * half;
        v2f a = *(const v2f*)&comb[m * 2 * DIMD + k0];
        v2f b = *(const v2f*)&sew[n * 2 * DIMD + k0];
        c = wmma_f32_4(a, b, c);
    }

    #pragma unroll
    for (int v = 0; v < 8; ++v)
        swt[wave][v + 8 * half][n] = c[v];
    __syncthreads();

    // serial per-row: bias -> leaky_relu(0.01) -> softmax ; zero-pad K=5..7
    if (lane < 16) {
        float L[NA], mx = -3.0e38f;
        #pragma unroll
        for (int k = 0; k < NA; ++k) {
            float l = swt[wave][lane][k] + sbias[k];
            l = (l > 0.0f) ? l : 0.01f * l;
            L[k] = l;
            mx = fmaxf(mx, l);
        }
        float s = 0.0f;
        #pragma unroll
        for (int k = 0; k < NA; ++k) { L[k] = __expf(L[k] - mx); s += L[k]; }
        float inv = 1.0f / s;
        #pragma unroll
        for (int k = 0; k < NA; ++k) swt[wave][lane][k] = L[k] * inv;
        #pragma unroll
        for (int k = NA; k < 8; ++k) swt[wave][lane][k] = 0.0f;
    }
    __syncthreads();

    // endpoint ids + validity for the rows this lane owns in the C/D layout
    int sidv[8], didv[8];
    float scale[8];
    #pragma unroll
    for (int v = 0; v < 8; ++v) {
        int mr = wave * 16 + v + 8 * half;
        sidv[v] = sid[mr];
        didv[v] = did[mr];
        // ids were clamped to valid rows; neutralize OOB edges by adding 0.0
        scale[v] = (ebase + mr < E) ? 0.5f : 0.0f;
    }

    // GEMM2: ep[16x128] = b[16x8pad] @ edge_anchor ; scatter 0.5*ep
    #pragma unroll
    for (int t8 = 0; t8 < DIMD / 16; ++t8) {
        v8f d = {};
        #pragma unroll
        for (int ks = 0; ks < 2; ++ks) {
            int k0 = 4 * ks + 2 * half;
            v2f a = *(const v2f*)&swt[wave][n][k0];
            v2f b;
            b.x = sea[(k0    ) * DIMD + t8 * 16 + n];
            b.y = sea[(k0 + 1) * DIMD + t8 * 16 + n];
            d = wmma_f32_4(a, b, d);
        }
        #pragma unroll
        for (int v = 0; v < 8; ++v) {
            float val = scale[v] * d[v];
            int col = t8 * 16 + n;
            atomicAdd(&out[(size_t)sidv[v] * DIMD + col], val);
            atomicAdd(&out[(size_t)didv[v] * DIMD + col], val);
        }
    }
}

extern "C" void kernel_launch(void* const* d_in, const int* in_sizes, int n_in,
                              void* d_out, int out_size, void* d_ws, size_t ws_size,
                              hipStream_t stream) {
    (void)n_in; (void)out_size; (void)d_ws; (void)ws_size;
    const float* x           = (const float*)d_in[0];
    const float* node_anchor = (const float*)d_in[1];
    const float* attn_w      = (const float*)d_in[2];
    const float* attn_b      = (const float*)d_in[3];
    const float* edge_anchor = (const float*)d_in[4];
    const float* edge_w      = (const float*)d_in[5];
    const float* edge_b      = (const float*)d_in[6];
    const int*   eidx        = (const int*)d_in[7];   // int64 data, read as int pairs
    float* out = (float*)d_out;

    const int N = in_sizes[0] / DIMD;
    const int E = in_sizes[7] / 2;

    // Pass 1: out = x + node prompt (fully overwrites d_out -> no init needed)
    node_prompt_kernel<<<(N + 63) / 64, 128, 0, stream>>>(
        x, attn_w, attn_b, node_anchor, out, N);
    // Pass 2: atomic scatter-add of 0.5 * edge prompt to both endpoints
    edge_prompt_kernel<<<(E + 31) / 32, 64, 0, stream>>>(
        x, edge_w, edge_b, edge_anchor, eidx, out, E);
}